// PCALayer_91010357002333
// MI455X (gfx1250) — compile-verified
//
#include <hip/hip_runtime.h>
#include <hip/hip_bf16.h>
#include <math.h>

typedef __attribute__((ext_vector_type(2))) float v2f;
typedef __attribute__((ext_vector_type(8))) float v8f;

#define B_DIM 8
#define C_DIM 64
#define N_DIM 65536        // H*W = 256*256
#define K_COMP 32

// ---------------------------------------------------------------------------
// Per-(b,c) mean over the N=65536 contiguous plane. One block per plane.
// ---------------------------------------------------------------------------
__global__ void pca_mean_kernel(const float* __restrict__ x, float* __restrict__ mean) {
    __shared__ float red[256];
    const int bc = blockIdx.x;                       // b*64 + c
    const float* plane = x + (size_t)bc * N_DIM;
    float s = 0.0f;
    for (int i = threadIdx.x; i < N_DIM; i += 256) s += plane[i];
    red[threadIdx.x] = s;
    __syncthreads();
    for (int off = 128; off > 0; off >>= 1) {
        if (threadIdx.x < (unsigned)off) red[threadIdx.x] += red[threadIdx.x + off];
        __syncthreads();
    }
    if (threadIdx.x == 0) mean[bc] = red[0] * (1.0f / (float)N_DIM);
}

// ---------------------------------------------------------------------------
// Gram[b] = X Xᵀ  (X is 64 x N, channel-major planes).  One block per
// (batch, 16x16 tile); 8 waves each contract an N/8 slice with fp32 WMMA
// (16x16x4), then deterministic LDS cross-wave reduce (no float atomics).
// ---------------------------------------------------------------------------
__global__ void pca_gram_kernel(const float* __restrict__ x, float* __restrict__ gram) {
    __shared__ float part[8 * 256];

    const int b       = blockIdx.x >> 4;
    const int tile    = blockIdx.x & 15;
    const int rowBase = (tile >> 2) << 4;
    const int colBase = (tile & 3) << 4;
    const int wave    = threadIdx.x >> 5;
    const int lane    = threadIdx.x & 31;
    const int lm      = lane & 15;
    const int dlt     = (lane < 16) ? 0 : 2;   // K offset held by this lane half

    // A tile: rows = channels rowBase..+15, K = spatial.  a.x -> K+dlt, a.y -> K+dlt+1
    const float* rowPlane = x + ((size_t)(b * C_DIM + rowBase + lm)) * N_DIM + dlt;
    // B tile: B[kk][n] = X[colBase+n][k+kk]
    const float* colPlane = x + ((size_t)(b * C_DIM + colBase + lm)) * N_DIM + dlt;

    const int kChunk = N_DIM / 8;
    const int k0 = wave * kChunk;
    const int k1 = k0 + kChunk;

    v8f acc0 = {};
    v8f acc1 = {};
    for (int k = k0; k < k1; k += 8) {
        v2f a0 = *(const v2f*)(rowPlane + k);
        v2f b0 = *(const v2f*)(colPlane + k);
        acc0 = __builtin_amdgcn_wmma_f32_16x16x4_f32(false, a0, false, b0,
                                                     (short)0, acc0, false, false);
        v2f a1 = *(const v2f*)(rowPlane + k + 4);
        v2f b1 = *(const v2f*)(colPlane + k + 4);
        acc1 = __builtin_amdgcn_wmma_f32_16x16x4_f32(false, a1, false, b1,
                                                     (short)0, acc1, false, false);
    }

#pragma unroll
    for (int i = 0; i < 8; ++i)
        part[wave * 256 + lane * 8 + i] = acc0[i] + acc1[i];
    __syncthreads();

    // Deterministic cross-wave reduction: thread t owns element (lane=t>>3, i=t&7)
    const int t = threadIdx.x;
    float s = 0.0f;
#pragma unroll
    for (int w = 0; w < 8; ++w) s += part[w * 256 + t];
    const int l2  = t >> 3;
    const int i2  = t & 7;
    const int row = rowBase + i2 + ((l2 < 16) ? 0 : 8);   // D layout: VGPR i -> M=i / i+8
    const int col = colBase + (l2 & 15);
    gram[(size_t)b * C_DIM * C_DIM + row * C_DIM + col] = s;
}

// ---------------------------------------------------------------------------
// Per-batch: cov = (Gram - N μμᵀ)/(N-1), cyclic Jacobi eigensolver on 64x64,
// select top-32 eigenvector columns by descending eigenvalue.
// One block of 64 threads per batch.
// ---------------------------------------------------------------------------
__global__ void pca_eig_kernel(const float* __restrict__ gram,
                               const float* __restrict__ mean,
                               float* __restrict__ topv) {
    __shared__ float A[C_DIM * 65];   // padded stride 65 (bank-conflict free)
    __shared__ float V[C_DIM * 65];
    __shared__ float cs[2];

    const int b = blockIdx.x;
    const int j = threadIdx.x;        // 0..63
    const float* g  = gram + (size_t)b * C_DIM * C_DIM;
    const float* mu = mean + b * C_DIM;
    const float mj  = mu[j];
    const float inv_nm1 = 1.0f / (float)(N_DIM - 1);

    for (int i = 0; i < C_DIM; ++i) {
        A[j * 65 + i] = (g[j * C_DIM + i] - (float)N_DIM * mj * mu[i]) * inv_nm1;
        V[j * 65 + i] = (i == j) ? 1.0f : 0.0f;
    }
    __syncthreads();

    for (int sweep = 0; sweep < 12; ++sweep) {
        for (int p = 0; p < C_DIM - 1; ++p) {
            for (int q = p + 1; q < C_DIM; ++q) {
                if (j == 0) {
                    float apq = A[p * 65 + q];
                    float app = A[p * 65 + p];
                    float aqq = A[q * 65 + q];
                    float c = 1.0f, s = 0.0f;
                    if (fabsf(apq) > 1e-12f) {
                        float theta = (aqq - app) / (2.0f * apq);
                        float tt = 1.0f / (fabsf(theta) + sqrtf(theta * theta + 1.0f));
                        if (theta < 0.0f) tt = -tt;
                        c = 1.0f / sqrtf(tt * tt + 1.0f);
                        s = tt * c;
                    }
                    cs[0] = c; cs[1] = s;
                }
                __syncthreads();
                const float c = cs[0], s = cs[1];
                // rotate rows p,q  (thread j owns column j)
                const float apj = A[p * 65 + j];
                const float aqj = A[q * 65 + j];
                A[p * 65 + j] = c * apj - s * aqj;
                A[q * 65 + j] = s * apj + c * aqj;
                __syncthreads();
                // rotate cols p,q  (thread j owns row j)
                const float ajp = A[j * 65 + p];
                const float ajq = A[j * 65 + q];
                A[j * 65 + p] = c * ajp - s * ajq;
                A[j * 65 + q] = s * ajp + c * ajq;
                const float vjp = V[j * 65 + p];
                const float vjq = V[j * 65 + q];
                V[j * 65 + p] = c * vjp - s * vjq;
                V[j * 65 + q] = s * vjp + c * vjq;
                __syncthreads();
            }
        }
    }

    // Rank eigenvalues descending; thread j owns eigenpair j.
    const float ev = A[j * 65 + j];
    int rank = 0;
    for (int i = 0; i < C_DIM; ++i) {
        const float evi = A[i * 65 + i];
        if (evi > ev || (evi == ev && i < j)) rank++;
    }
    if (rank < K_COMP) {
        float* tv = topv + (size_t)b * C_DIM * K_COMP;
        for (int cc = 0; cc < C_DIM; ++cc)
            tv[cc * K_COMP + rank] = V[cc * 65 + j];   // topv[c][rank] = V[c][j]
    }
}

// ---------------------------------------------------------------------------
// Projection: out[b,k,n] = Σ_c (x[b,c,n]-μ[b,c]) * topv[c,k].
// D (16k x 16n) = A(Vᵀ tile 16x4) * B(centered 4x16), 16 fp32 WMMAs over C=64.
// Grid (1024, 8) blocks x 256 threads; each wave owns one (k-tile, n-subtile).
// ---------------------------------------------------------------------------
__global__ void pca_proj_kernel(const float* __restrict__ x,
                                const float* __restrict__ mean,
                                const float* __restrict__ topv,
                                float* __restrict__ out) {
    __shared__ float smu[C_DIM];
    const int b = blockIdx.y;
    if (threadIdx.x < C_DIM) smu[threadIdx.x] = mean[b * C_DIM + threadIdx.x];
    __syncthreads();

    const int wave  = threadIdx.x >> 5;
    const int lane  = threadIdx.x & 31;
    const int lm    = lane & 15;
    const int dlt   = (lane < 16) ? 0 : 2;
    const int kBase = (wave & 1) << 4;                        // 0 or 16
    const int nBase = (blockIdx.x * 4 + (wave >> 1)) << 4;    // 16-wide spatial tile

    const float* tv = topv + (size_t)b * C_DIM * K_COMP;
    const float* xb = x + (size_t)b * C_DIM * N_DIM;

    v8f acc = {};
    for (int c = 0; c < C_DIM; c += 4) {
        const int c0 = c + dlt;
        v2f a;   // A[m=k, kk=c]: a.x = V[c0][kBase+lm], a.y = V[c0+1][kBase+lm]
        a.x = tv[c0 * K_COMP + kBase + lm];
        a.y = tv[(c0 + 1) * K_COMP + kBase + lm];
        v2f bm;  // B[kk=c, n]: centered samples, coalesced 16-float runs per half-wave
        bm.x = xb[(size_t)c0 * N_DIM + nBase + lm] - smu[c0];
        bm.y = xb[(size_t)(c0 + 1) * N_DIM + nBase + lm] - smu[c0 + 1];
        acc = __builtin_amdgcn_wmma_f32_16x16x4_f32(false, a, false, bm,
                                                    (short)0, acc, false, false);
    }

    float* ob = out + (size_t)b * K_COMP * N_DIM;
    const int k0 = kBase + ((lane < 16) ? 0 : 8);
#pragma unroll
    for (int i = 0; i < 8; ++i)
        ob[(size_t)(k0 + i) * N_DIM + nBase + lm] = acc[i];   // coalesced along n
}

// ---------------------------------------------------------------------------
extern "C" void kernel_launch(void* const* d_in, const int* in_sizes, int n_in,
                              void* d_out, int out_size, void* d_ws, size_t ws_size,
                              hipStream_t stream) {
    (void)in_sizes; (void)n_in; (void)out_size; (void)ws_size;
    const float* x = (const float*)d_in[0];   // [8,64,256,256] fp32
    float* out = (float*)d_out;               // [8,32,256,256] fp32

    // Workspace layout (floats): mean[512] | gram[8*64*64] | topv[8*64*32]
    float* mean = (float*)d_ws;
    float* gram = mean + B_DIM * C_DIM;
    float* topv = gram + B_DIM * C_DIM * C_DIM;

    pca_mean_kernel<<<B_DIM * C_DIM, 256, 0, stream>>>(x, mean);
    pca_gram_kernel<<<B_DIM * 16, 256, 0, stream>>>(x, gram);
    pca_eig_kernel<<<B_DIM, C_DIM, 0, stream>>>(gram, mean, topv);
    pca_proj_kernel<<<dim3(N_DIM / 64, B_DIM), 256, 0, stream>>>(x, mean, topv, out);
}